// GCN_28329604284476
// MI455X (gfx1250) — compile-verified
//
#include <hip/hip_runtime.h>
#include <hip/hip_bf16.h>

typedef __attribute__((ext_vector_type(2))) float v2f;
typedef __attribute__((ext_vector_type(8))) float v8f;

// ---------------------------------------------------------------------------
// Degree / normalization kernels
// ---------------------------------------------------------------------------
__global__ void deg_init_kernel(float* __restrict__ deg, int n) {
  int i = blockIdx.x * blockDim.x + threadIdx.x;
  if (i < n) deg[i] = 1.0f;  // self loop contributes +1
}

__global__ void deg_count_kernel(const int* __restrict__ dst, float* __restrict__ deg, int E) {
  int e = blockIdx.x * blockDim.x + threadIdx.x;
  if (e < E) atomicAdd(&deg[dst[e]], 1.0f);
}

__global__ void dinv_kernel(float* __restrict__ deg, int n) {
  int i = blockIdx.x * blockDim.x + threadIdx.x;
  if (i < n) deg[i] = rsqrtf(deg[i]);  // deg becomes dinv in place
}

// ---------------------------------------------------------------------------
// fp32 WMMA GEMM: C[M x N] = A[M x K] * B[K x N]
// One wave computes one 16x16 tile with V_WMMA_F32_16X16X4_F32.
// Requires: M % 16 == 0, N % 16 == 0, K % 4 == 0 (true for this workload).
// A 16x4 layout : lanes 0-15 = rows, VGPR{0,1} = K {0,1} (lanes 0-15) / {2,3} (16-31)
// B 4x16 layout : lanes 0-15 = cols, VGPR{0,1} = K {0,1} (lanes 0-15) / {2,3} (16-31)
// C 16x16 layout: VGPR v -> row v (lanes 0-15) / row v+8 (lanes 16-31), col = lane%16
// ---------------------------------------------------------------------------
__global__ void wmma_gemm_f32_kernel(const float* __restrict__ A,
                                     const float* __restrict__ B,
                                     float* __restrict__ C,
                                     int M, int K, int N) {
  const int wave = (blockIdx.x * blockDim.x + threadIdx.x) >> 5;
  const int lane = threadIdx.x & 31;
  const int ntiles = N >> 4;
  const int row0 = (wave / ntiles) << 4;
  const int col0 = (wave % ntiles) << 4;
  if (row0 >= M) return;  // wave-uniform: EXEC stays all-ones for WMMA

  const int half = lane >> 4;   // 0: lanes 0-15, 1: lanes 16-31
  const int r    = lane & 15;

  v8f acc = {};
  const float* __restrict__ arow = A + (size_t)(row0 + r) * K;

#pragma unroll 4
  for (int k0 = 0; k0 < K; k0 += 4) {
    const int ka = k0 + (half << 1);
    v2f a, b;
    a.x = arow[ka];
    a.y = arow[ka + 1];
    b.x = B[(size_t)ka * N + col0 + r];
    b.y = B[(size_t)(ka + 1) * N + col0 + r];
    // (neg_a, A, neg_b, B, c_mod, C, reuse_a, reuse_b)
    acc = __builtin_amdgcn_wmma_f32_16x16x4_f32(false, a, false, b,
                                                (short)0, acc, false, false);
  }

  float* __restrict__ crow = C + (size_t)(row0 + (half << 3)) * N + col0 + r;
#pragma unroll
  for (int v = 0; v < 8; ++v) crow[(size_t)v * N] = acc[v];
}

// ---------------------------------------------------------------------------
// agg[i][f] = h[i][f] * dinv[i]^2   (zero-init + self-loop term in one pass)
// ---------------------------------------------------------------------------
template <int LOGF>
__global__ void self_loop_init_kernel(const float* __restrict__ h,
                                      const float* __restrict__ dinv,
                                      float* __restrict__ agg,
                                      unsigned total) {
  unsigned t = blockIdx.x * blockDim.x + threadIdx.x;
  if (t >= total) return;
  unsigned i = t >> LOGF;
  float di = dinv[i];
  agg[t] = h[t] * (di * di);
}

// ---------------------------------------------------------------------------
// Edge scatter: agg[dst][f] += h[src][f] * dinv[src]*dinv[dst]
// thread = (edge, feature); consecutive threads -> consecutive features,
// so gather loads and L2 atomics are fully coalesced per edge row.
// ---------------------------------------------------------------------------
template <int LOGF>
__global__ void edge_scatter_kernel(const float* __restrict__ h,
                                    const int* __restrict__ src,
                                    const int* __restrict__ dst,
                                    const float* __restrict__ dinv,
                                    float* __restrict__ agg,
                                    unsigned total) {
  unsigned t = blockIdx.x * blockDim.x + threadIdx.x;
  if (t >= total) return;
  const unsigned F = 1u << LOGF;
  unsigned e = t >> LOGF;
  unsigned f = t & (F - 1u);
  int s = src[e];
  int d = dst[e];
  float norm = dinv[s] * dinv[d];
  atomicAdd(&agg[(size_t)d * F + f], h[(size_t)s * F + f] * norm);
}

// ---------------------------------------------------------------------------
// x = relu(x + bias[f])  in place
// ---------------------------------------------------------------------------
template <int LOGF>
__global__ void bias_relu_kernel(float* __restrict__ x,
                                 const float* __restrict__ bias,
                                 unsigned total) {
  unsigned t = blockIdx.x * blockDim.x + threadIdx.x;
  if (t >= total) return;
  const unsigned F = 1u << LOGF;
  float v = x[t] + bias[t & (F - 1u)];
  x[t] = fmaxf(v, 0.0f);
}

// ---------------------------------------------------------------------------
// Launch
// ---------------------------------------------------------------------------
extern "C" void kernel_launch(void* const* d_in, const int* in_sizes, int n_in,
                              void* d_out, int out_size, void* d_ws, size_t ws_size,
                              hipStream_t stream) {
  const float* X  = (const float*)d_in[0];   // [N, 256]
  const int*   ei = (const int*)d_in[1];     // [2, E]
  const float* W1 = (const float*)d_in[2];   // [256, 64]
  const float* b1 = (const float*)d_in[3];   // [64]
  const float* W2 = (const float*)d_in[4];   // [64, 256]
  const float* b2 = (const float*)d_in[5];   // [256]
  float* out = (float*)d_out;                // [N, 256]

  const int N = in_sizes[0] / 256;           // 50000
  const int E = in_sizes[1] / 2;             // 800000
  const int* src = ei;
  const int* dst = ei + E;

  // Workspace layout
  char* p = (char*)d_ws;
  float* dinv  = (float*)p;  p += (size_t)N * sizeof(float);          // deg -> dinv
  float* h1lin = (float*)p;  p += (size_t)N * 64 * sizeof(float);
  float* agg1  = (float*)p;  p += (size_t)N * 64 * sizeof(float);     // becomes h1
  float* h2lin = (float*)p;  p += (size_t)N * 256 * sizeof(float);

  const int TB = 256;
  auto blocks = [](unsigned work, unsigned tb) { return (work + tb - 1) / tb; };

  // --- degree / dinv ---
  deg_init_kernel<<<blocks(N, TB), TB, 0, stream>>>(dinv, N);
  deg_count_kernel<<<blocks(E, TB), TB, 0, stream>>>(dst, dinv, E);
  dinv_kernel<<<blocks(N, TB), TB, 0, stream>>>(dinv, N);

  // --- layer 1: h1lin = X @ W1  (M=N, K=256, Nout=64) ---
  {
    unsigned waves = (unsigned)(N / 16) * (64 / 16);
    wmma_gemm_f32_kernel<<<blocks(waves * 32u, TB), TB, 0, stream>>>(X, W1, h1lin, N, 256, 64);
  }
  {
    unsigned total = (unsigned)N * 64u;
    self_loop_init_kernel<6><<<blocks(total, TB), TB, 0, stream>>>(h1lin, dinv, agg1, total);
  }
  {
    unsigned total = (unsigned)E * 64u;
    edge_scatter_kernel<6><<<blocks(total, TB), TB, 0, stream>>>(h1lin, src, dst, dinv, agg1, total);
  }
  {
    unsigned total = (unsigned)N * 64u;
    bias_relu_kernel<6><<<blocks(total, TB), TB, 0, stream>>>(agg1, b1, total);  // agg1 = h1
  }

  // --- layer 2: h2lin = h1 @ W2  (M=N, K=64, Nout=256) ---
  {
    unsigned waves = (unsigned)(N / 16) * (256 / 16);
    wmma_gemm_f32_kernel<<<blocks(waves * 32u, TB), TB, 0, stream>>>(agg1, W2, h2lin, N, 64, 256);
  }
  {
    unsigned total = (unsigned)N * 256u;
    self_loop_init_kernel<8><<<blocks(total, TB), TB, 0, stream>>>(h2lin, dinv, out, total);
  }
  {
    unsigned total = (unsigned)E * 256u;
    edge_scatter_kernel<8><<<blocks(total, TB), TB, 0, stream>>>(h2lin, src, dst, dinv, out, total);
  }
  {
    unsigned total = (unsigned)N * 256u;
    bias_relu_kernel<8><<<blocks(total, TB), TB, 0, stream>>>(out, b2, total);
  }
}